// SSPUF_ODE_64037962383820
// MI455X (gfx1250) — compile-verified
//
#include <hip/hip_runtime.h>
#include <hip/hip_bf16.h>

// ---------------------------------------------------------------------------
// SSPUF_ODE: two LC-star state matrices -> truncated expm series (24 dense
// fp32 GEMMs per star, done with V_WMMA_F32_16X16X4_F32) -> 64 ZOH steps.
// Padded state dim: 1025 -> NP = 1056 (= 66*16 = 33*32 = 11*96).
// GEMM grid is chosen so every wave covers exactly in-range tiles: no guards,
// branchless WMMA inner loop, EXEC all-1s throughout (WMMA requirement).
// ---------------------------------------------------------------------------

#define NPAD 1056
#define NREAL 1025
#define MIDI 1024
#define NORDER 24
#define NSTEPS 64

typedef __attribute__((ext_vector_type(2))) float v2f;
typedef __attribute__((ext_vector_type(8))) float v8f;

// --------------------------- init: Adt=0, T=I, E=I, Phi=I*dt, x=0 ----------
__global__ void sspuf_init_kernel(float* __restrict__ Adt, float* __restrict__ Ta,
                                  float* __restrict__ E, float* __restrict__ Phi,
                                  float* __restrict__ xa, const float* __restrict__ t_ptr)
{
    const size_t M2 = (size_t)NPAD * NPAD;
    size_t idx = (size_t)blockIdx.x * blockDim.x + threadIdx.x;
    float dt = t_ptr[0] * (1.0f / (float)NSTEPS);
    if (idx < 2 * M2) {
        size_t e = idx % M2;
        int r = (int)(e / NPAD), c = (int)(e % NPAD);
        float diag = (r == c && r < NREAL) ? 1.0f : 0.0f;
        Adt[idx] = 0.0f;
        Ta[idx]  = diag;
        E[idx]   = diag;
        Phi[idx] = diag * dt;
    }
    if (idx < 2 * NPAD) xa[idx] = 0.0f;
}

// --------------------------- build A*dt (sparse scatter) -------------------
// One block per star, 512 threads = (branch b in 0..15) x (k in 0..31).
__global__ void sspuf_build_kernel(const int* __restrict__ sw, const float* __restrict__ mm,
                                   const float* __restrict__ gmc_p, const float* __restrict__ gml_p,
                                   const float* __restrict__ cval, const float* __restrict__ lval,
                                   const float* __restrict__ t_ptr, float* __restrict__ Adt)
{
    const int L = 32, Bn = 16;
    int s   = blockIdx.x;
    int tid = threadIdx.x;          // 0..511
    int b   = tid >> 5;
    int k   = tid & 31;
    const float* m = mm + (size_t)s * 3073 + (size_t)b * 192;
    float dt   = t_ptr[0] * (1.0f / (float)NSTEPS);
    float gmc0 = m[2 * k]     * gmc_p[2 * k];
    float gmc1 = m[2 * k + 1] * gmc_p[2 * k + 1];
    float gml0 = m[64 + 2 * k]     * gml_p[2 * k];
    float gml1 = m[64 + 2 * k + 1] * gml_p[2 * k + 1];
    float C    = 1e-9f * m[128 + k] * cval[k];
    float Lm   = 1e-9f * m[160 + k] * lval[k];
    float swb  = (float)sw[b];
    float* A = Adt + (size_t)s * NPAD * NPAD;
    int vi = b * L + k;
    int ii = Bn * L + b * L + k;
    // dv/dt rows
    A[(size_t)vi * NPAD + ii] = dt * gmc0 / C;
    if (k < L - 1) A[(size_t)vi * NPAD + ii + 1] = -dt * gmc1 / C;
    // di/dt rows
    if (k >= 1) A[(size_t)ii * NPAD + vi - 1] = dt * gml0 / Lm;
    if (k == 0) A[(size_t)ii * NPAD + MIDI]   = dt * swb * gml0 / Lm;
    A[(size_t)ii * NPAD + vi] = -dt * gml1 / Lm;
    // dv_mid/dt row
    if (k == 0) {
        float cmid = 1e-9f * mm[(size_t)s * 3073 + 3072] * cval[L];
        A[(size_t)MIDI * NPAD + ii] = -dt * swb / cmid;
    }
}

// --------------------------- fp32 WMMA GEMM step ---------------------------
// Tnew = (Told @ Adt) * (1/j);  E += Tnew;  Phi += Tnew * dt/(j+1)
// grid = (11 N-strips of 96, 11 M-groups, 2 stars), block = 192 (6 waves).
// Each wave: one 16-row M tile x 96-col N strip (6 accumulators), K-step 4.
// 11*6 waves cover 66 M tiles exactly; 11 strips * 6 tiles cover 66 N tiles
// exactly -> completely branchless inner loop, no bounds guards.
__global__ void sspuf_gemm_kernel(const float* __restrict__ Told, const float* __restrict__ Adt,
                                  float* __restrict__ Tnew, float* __restrict__ E,
                                  float* __restrict__ Phi, const float* __restrict__ t_ptr, int j)
{
    const size_t M2 = (size_t)NPAD * NPAD;
    int s    = blockIdx.z;
    int wave = threadIdx.x >> 5;          // 0..5 (wave32)
    int lane = threadIdx.x & 31;
    int mt   = blockIdx.y * 6 + wave;     // 0..65, exact cover: no partial waves
    int m0   = mt * 16;
    int n0   = blockIdx.x * 96;           // 0..960, exact cover
    size_t base = (size_t)s * M2;
    const float* A = Told + base;
    const float* B = Adt + base;

    int row   = m0 + (lane & 15);         // A-frag: M row per lane
    int khalf = lane >> 4;                // 0: K={0,1}, 1: K={2,3}
    int coln  = lane & 15;                // B/C-frag: N column per lane

    v8f z = {0.f, 0.f, 0.f, 0.f, 0.f, 0.f, 0.f, 0.f};
    v8f acc[6] = {z, z, z, z, z, z};

    const float* arow = A + (size_t)row * NPAD + khalf * 2;
    const float* bcol = B + (size_t)(khalf * 2) * NPAD + n0 + coln;

    for (int k = 0; k < NPAD; k += 4) {
        // A fragment (16x4 fp32): 2 contiguous floats per lane
        v2f a;
        a.x = arow[k];
        a.y = arow[k + 1];
        const float* bk = bcol + (size_t)k * NPAD;
        #pragma unroll
        for (int t6 = 0; t6 < 6; ++t6) {
            v2f b;
            b.x = bk[t6 * 16];
            b.y = bk[t6 * 16 + NPAD];
            acc[t6] = __builtin_amdgcn_wmma_f32_16x16x4_f32(
                false, a, false, b, (short)0, acc[t6], false, false);
        }
    }

    float dt   = t_ptr[0] * (1.0f / (float)NSTEPS);
    float invj = 1.0f / (float)j;
    float cphi = dt / (float)(j + 1);
    // C/D layout: VGPR r -> M = m0 + 8*khalf + r, N = n0 + t6*16 + coln
    #pragma unroll
    for (int t6 = 0; t6 < 6; ++t6) {
        int n = n0 + t6 * 16;
        #pragma unroll
        for (int r = 0; r < 8; ++r) {
            int rr = m0 + khalf * 8 + r;
            size_t off = base + (size_t)rr * NPAD + n + coln;
            float v = acc[t6][r] * invj;
            Tnew[off] = v;
            E[off]   += v;
            Phi[off] += v * cphi;
        }
    }
}

// --------------------------- phib = Phi @ Bv = Phi[:,mid] / c_mid ----------
__global__ void sspuf_phib_kernel(const float* __restrict__ Phi, const float* __restrict__ mm,
                                  const float* __restrict__ cval, float* __restrict__ phib)
{
    int idx = blockIdx.x * blockDim.x + threadIdx.x;
    if (idx >= 2 * NPAD) return;
    int s = idx / NPAD, i = idx % NPAD;
    float cmid = 1e-9f * mm[(size_t)s * 3073 + 3072] * cval[32];
    phib[idx] = Phi[(size_t)s * NPAD * NPAD + (size_t)i * NPAD + MIDI] / cmid;
}

// --------------------------- trapezoidal pulse -----------------------------
__device__ __forceinline__ float sspuf_pulse(float t)
{
    const float rise = 5e-10f, width = 1e-9f, fall = 5e-10f;
    if (t < rise) return t * (1.0f / rise);
    if (t < rise + width) return 1.0f;
    if (t < rise + width + fall) return 1.0f - (t - rise - width) * (1.0f / fall);
    return 0.0f;
}

// --------------------------- one ZOH step: x' = E x + phib*u ---------------
// grid = (132 row-groups of 8, 2 stars), block = 256 (8 waves, wave per row).
__global__ void sspuf_step_kernel(const float* __restrict__ E, const float* __restrict__ xin,
                                  float* __restrict__ xout, const float* __restrict__ phib,
                                  const float* __restrict__ t_ptr, int kstep)
{
    int s    = blockIdx.y;
    int wave = threadIdx.x >> 5;
    int lane = threadIdx.x & 31;
    int row  = blockIdx.x * 8 + wave;   // 132*8 = 1056 exact
    const float* Em = E + (size_t)s * NPAD * NPAD + (size_t)row * NPAD;
    const float* xv = xin + (size_t)s * NPAD;
    float sum = 0.0f;
    for (int j = lane; j < NPAD; j += 32)
        sum += Em[j] * xv[j];
    #pragma unroll
    for (int off = 16; off > 0; off >>= 1)
        sum += __shfl_xor(sum, off, 32);
    if (lane == 0) {
        float dt = t_ptr[0] * (1.0f / (float)NSTEPS);
        float u  = sspuf_pulse(((float)kstep + 0.5f) * dt);
        xout[(size_t)s * NPAD + row] = sum + phib[(size_t)s * NPAD + row] * u;
    }
}

// --------------------------- readout ---------------------------------------
__global__ void sspuf_final_kernel(const float* __restrict__ x, float* __restrict__ out)
{
    out[0] = x[MIDI] - x[NPAD + MIDI];
}

// ---------------------------------------------------------------------------
extern "C" void kernel_launch(void* const* d_in, const int* in_sizes, int n_in,
                              void* d_out, int out_size, void* d_ws, size_t ws_size,
                              hipStream_t stream)
{
    (void)in_sizes; (void)n_in; (void)out_size; (void)ws_size;
    const size_t M2 = (size_t)NPAD * NPAD;

    const int*   sw    = (const int*)d_in[0];
    const float* mm    = (const float*)d_in[1];
    const float* gmc   = (const float*)d_in[2];
    const float* gml   = (const float*)d_in[3];
    const float* cval  = (const float*)d_in[4];
    const float* lval  = (const float*)d_in[5];
    const float* t_ptr = (const float*)d_in[6];
    float* out = (float*)d_out;

    float* W    = (float*)d_ws;
    float* Adt  = W;
    float* Ta   = W + 2 * M2;
    float* Tb   = W + 4 * M2;
    float* E    = W + 6 * M2;
    float* Phi  = W + 8 * M2;
    float* phib = W + 10 * M2;
    float* xa   = phib + 2 * NPAD;
    float* xb   = xa + 2 * NPAD;

    unsigned nb = (unsigned)((2 * M2 + 255) / 256);
    sspuf_init_kernel<<<nb, 256, 0, stream>>>(Adt, Ta, E, Phi, xa, t_ptr);
    sspuf_build_kernel<<<2, 512, 0, stream>>>(sw, mm, gmc, gml, cval, lval, t_ptr, Adt);

    float* Told = Ta; float* Tnew = Tb;
    for (int j = 1; j <= NORDER; ++j) {
        sspuf_gemm_kernel<<<dim3(11, 11, 2), 192, 0, stream>>>(Told, Adt, Tnew, E, Phi, t_ptr, j);
        float* tmp = Told; Told = Tnew; Tnew = tmp;
    }

    sspuf_phib_kernel<<<(2 * NPAD + 255) / 256, 256, 0, stream>>>(Phi, mm, cval, phib);

    float* xin = xa; float* xout = xb;
    for (int k = 0; k < NSTEPS; ++k) {
        sspuf_step_kernel<<<dim3(132, 2), 256, 0, stream>>>(E, xin, xout, phib, t_ptr, k);
        float* tmp = xin; xin = xout; xout = tmp;
    }

    sspuf_final_kernel<<<1, 1, 0, stream>>>(xin, out);
}